// InhibitionMemoryLayer_13134009991920
// MI455X (gfx1250) — compile-verified
//
#include <hip/hip_runtime.h>
#include <math.h>

#define HIDDEN 768
#define INPUT  256
#define BATCH  32
#define TSTEPS 128
#define TWOH   1536

// exp(-1/20) for both tau_mem and tau_trace
#define ALPHA_C   0.95122942450071400910f
#define OMA_C     0.04877057549928599090f
#define DECAY_C   0.95122942450071400910f
#define OMD_C     0.04877057549928599090f
#define ETA_C     0.1f
#define IKV_SCALE 0.02f     // 0.2 * ETA (mem folded out of the scan)

typedef float v2f __attribute__((ext_vector_type(2)));
typedef float v8f __attribute__((ext_vector_type(8)));

// ---------------------------------------------------------------------------
// Kernel 1: I[m,n] = sum_k X[m,k] * W[n,k]   (M=4096, N=1536, K=256)
// One wave per 16x16 C tile via V_WMMA_F32_16X16X4_F32. Both operands are
// k-contiguous -> one b64 load per lane per fragment.
// ---------------------------------------------------------------------------
__global__ __launch_bounds__(256)
void gemm_xw_wmma(const float* __restrict__ X,
                  const float* __restrict__ W,
                  float* __restrict__ I) {
    const int lane = threadIdx.x & 31;
    const int wave = threadIdx.x >> 5;
    const int m0   = blockIdx.y * 16;
    const int n0   = (blockIdx.x * 8 + wave) * 16;
    const int half = lane >> 4;
    const int l16  = lane & 15;

    const float* ap = X + (size_t)(m0 + l16) * INPUT + half * 2;
    const float* bp = W + (size_t)(n0 + l16) * INPUT + half * 2;

    v8f c = {};
#pragma unroll 8
    for (int k = 0; k < INPUT; k += 4) {
        v2f a = *(const v2f*)(ap + k);
        v2f b = *(const v2f*)(bp + k);
        c = __builtin_amdgcn_wmma_f32_16x16x4_f32(false, a, false, b,
                                                  (short)0, c, false, false);
    }

    const int mbase = m0 + half * 8;
#pragma unroll
    for (int r = 0; r < 8; ++r) {
        I[(size_t)(mbase + r) * TWOH + n0 + l16] = c[r];
    }
}

// ---------------------------------------------------------------------------
// Kernel 2: key path (mem-independent). One thread per (b,h).
// Writes key_o to the output and the ktr trace to workspace.
// ---------------------------------------------------------------------------
__global__ __launch_bounds__(256)
void key_scan(const float* __restrict__ I,
              float* __restrict__ keys_out,
              float* __restrict__ ktr_ws) {
    const int idx = blockIdx.x * blockDim.x + threadIdx.x;   // [0, B*H)
    const int b = idx / HIDDEN;
    const int h = idx - b * HIDDEN;

    const float* ik = I + (size_t)b * TSTEPS * TWOH + h;
    float* ko = keys_out + (size_t)b * TSTEPS * HIDDEN + h;
    float* kt = ktr_ws   + (size_t)b * TSTEPS * HIDDEN + h;

    float kv = 0.f, ktr = 0.f;
    for (int t = 0; t < TSTEPS; ++t) {
        kv = ALPHA_C * kv + OMA_C * ik[(size_t)t * TWOH];
        float key = tanhf(kv);
        ktr = DECAY_C * ktr + OMD_C * key;
        ko[(size_t)t * HIDDEN] = key;
        kt[(size_t)t * HIDDEN] = ktr;
    }
}

// ---------------------------------------------------------------------------
// Kernel 3: per-batch Gram matrix C[s,t] = sum_h ktr[b,s,h]*key[b,t,h]
// (M=N=T=128, K=768). Both operands k(h)-contiguous -> b64 fragment loads.
// grid: (8 s-tiles, 32 batches); 8 waves cover all 8 t-tiles.
// ---------------------------------------------------------------------------
__global__ __launch_bounds__(256)
void gram_wmma(const float* __restrict__ ktr_ws,
               const float* __restrict__ keys,
               float* __restrict__ Cws) {
    const int lane = threadIdx.x & 31;
    const int wave = threadIdx.x >> 5;
    const int b    = blockIdx.y;
    const int s0   = blockIdx.x * 16;
    const int t0   = wave * 16;
    const int half = lane >> 4;
    const int l16  = lane & 15;

    const float* ap = ktr_ws + (size_t)b * TSTEPS * HIDDEN
                      + (size_t)(s0 + l16) * HIDDEN + half * 2;
    const float* bp = keys   + (size_t)b * TSTEPS * HIDDEN
                      + (size_t)(t0 + l16) * HIDDEN + half * 2;

    v8f c = {};
#pragma unroll 8
    for (int k = 0; k < HIDDEN; k += 4) {
        v2f a = *(const v2f*)(ap + k);
        v2f b = *(const v2f*)(bp + k);
        c = __builtin_amdgcn_wmma_f32_16x16x4_f32(false, a, false, b,
                                                  (short)0, c, false, false);
    }

    float* co = Cws + (size_t)b * TSTEPS * TSTEPS;
    const int sbase = s0 + half * 8;
#pragma unroll
    for (int r = 0; r < 8; ++r) {
        co[(size_t)(sbase + r) * TSTEPS + t0 + l16] = c[r];
    }
}

// ---------------------------------------------------------------------------
// Kernel 4: value scan with mem folded out.
//   ikv_t[i] = 0.02 * sum_{s<t} vtr_s[i] * C[s,t]
// Each thread owns one i: its vtr history is a private LDS column (stride 257
// -> conflict-free: bank = (s + tid) % 64), C[s,t] is a lane-uniform LDS
// broadcast. NO barriers inside the T loop. Ends by writing vtr transposed
// (t-contiguous) for the final mem GEMM.
// ---------------------------------------------------------------------------
#define VSTRIDE 257

__global__ __launch_bounds__(256)
void value_scan(const float* __restrict__ I,
                const float* __restrict__ Cws,
                float* __restrict__ vals_out,
                float* __restrict__ vtr_ws) {
    extern __shared__ float smem[];
    float* cS   = smem;                        // TSTEPS*TSTEPS = 16384
    float* vtrH = cS + TSTEPS * TSTEPS;        // TSTEPS*VSTRIDE = 32896

    const int tid = threadIdx.x;
    const int b   = blockIdx.y;
    const int i0  = blockIdx.x * 256;

    const float* cg = Cws + (size_t)b * TSTEPS * TSTEPS;
    for (int idx = tid; idx < TSTEPS * TSTEPS; idx += 256) cS[idx] = cg[idx];
    __syncthreads();

    const float* ivp = I + (size_t)b * TSTEPS * TWOH + HIDDEN + i0;
    float* vout = vals_out + (size_t)b * TSTEPS * HIDDEN + i0;

    float vv = 0.f, vtr = 0.f;
    for (int t = 0; t < TSTEPS; ++t) {
        float acc = 0.f;
        for (int s = 0; s < t; ++s)
            acc += vtrH[s * VSTRIDE + tid] * cS[s * TSTEPS + t];
        float ikv = IKV_SCALE * acc;
        float iv  = ivp[(size_t)t * TWOH + tid];
        vv = ALPHA_C * vv + OMA_C * (iv + ikv);
        float val = tanhf(vv);
        vout[(size_t)t * HIDDEN + tid] = val;
        vtr = DECAY_C * vtr + OMD_C * val;
        vtrH[t * VSTRIDE + tid] = vtr;          // own column only
    }
    __syncthreads();

    // transpose-out vtr history: vtr_ws[b][i][t], t-contiguous, coalesced
    float* vw = vtr_ws + ((size_t)b * HIDDEN + i0) * TSTEPS;
    for (int idx = tid; idx < 256 * TSTEPS; idx += 256) {
        int il = idx >> 7;          // 0..255
        int t  = idx & 127;
        vw[(size_t)il * TSTEPS + t] = vtrH[t * VSTRIDE + il];
    }
}

// ---------------------------------------------------------------------------
// Kernel 5: mem[b,i,j] = ETA * sum_t vtr[b,i,t] * ktr[b,t,j]
// (M=N=768, K=128 per batch). A is k(t)-contiguous (b64 loads); B (ktr) is
// K-major -> two coalesced b32 loads per fragment.
// ---------------------------------------------------------------------------
__global__ __launch_bounds__(256)
void mem_wmma(const float* __restrict__ vtr_ws,
              const float* __restrict__ ktr_ws,
              float* __restrict__ mem_out) {
    const int lane = threadIdx.x & 31;
    const int wave = threadIdx.x >> 5;
    const int b    = blockIdx.z;
    const int m0   = blockIdx.y * 16;                 // 48 i-tiles
    const int n0   = (blockIdx.x * 8 + wave) * 16;    // 6 blocks * 8 waves
    const int half = lane >> 4;
    const int l16  = lane & 15;

    const float* ap = vtr_ws + ((size_t)b * HIDDEN + m0 + l16) * TSTEPS + half * 2;
    const float* bp = ktr_ws + (size_t)b * TSTEPS * HIDDEN
                      + (size_t)(half * 2) * HIDDEN + n0 + l16;

    v8f c = {};
#pragma unroll 8
    for (int k = 0; k < TSTEPS; k += 4) {
        v2f a = *(const v2f*)(ap + k);
        v2f bfrag;
        bfrag.x = bp[(size_t)k * HIDDEN];             // B[k+2*half][n]
        bfrag.y = bp[(size_t)(k + 1) * HIDDEN];       // B[k+2*half+1][n]
        c = __builtin_amdgcn_wmma_f32_16x16x4_f32(false, a, false, bfrag,
                                                  (short)0, c, false, false);
    }

    float* mo = mem_out + (size_t)b * HIDDEN * HIDDEN;
    const int mbase = m0 + half * 8;
#pragma unroll
    for (int r = 0; r < 8; ++r) {
        mo[(size_t)(mbase + r) * HIDDEN + n0 + l16] = ETA_C * c[r];
    }
}

// ---------------------------------------------------------------------------
extern "C" void kernel_launch(void* const* d_in, const int* in_sizes, int n_in,
                              void* d_out, int out_size, void* d_ws, size_t ws_size,
                              hipStream_t stream) {
    const float* x = (const float*)d_in[0];   // (B,T,INPUT)
    const float* W = (const float*)d_in[1];   // (2H, INPUT)

    float* ws_i   = (float*)d_ws;                                   // B*T*2H
    float* ws_ktr = ws_i   + (size_t)BATCH * TSTEPS * TWOH;         // B*T*H
    float* ws_C   = ws_ktr + (size_t)BATCH * TSTEPS * HIDDEN;       // B*T*T
    float* ws_vtr = ws_C   + (size_t)BATCH * TSTEPS * TSTEPS;       // B*H*T

    float* out_mem  = (float*)d_out;                                // B*H*H
    float* out_keys = out_mem  + (size_t)BATCH * HIDDEN * HIDDEN;   // B*T*H
    float* out_vals = out_keys + (size_t)BATCH * TSTEPS * HIDDEN;   // B*T*H

    // 1) input projection GEMM (fp32 WMMA)
    dim3 g1(TWOH / 128, (BATCH * TSTEPS) / 16);
    gemm_xw_wmma<<<g1, 256, 0, stream>>>(x, W, ws_i);

    // 2) key-path scan
    key_scan<<<(BATCH * HIDDEN) / 256, 256, 0, stream>>>(ws_i, out_keys, ws_ktr);

    // 3) per-batch Gram matrix C = Ktr * Keys^T (fp32 WMMA)
    dim3 g3(TSTEPS / 16, BATCH);
    gram_wmma<<<g3, 256, 0, stream>>>(ws_ktr, out_keys, ws_C);

    // 4) value scan (triangular, barrier-free inner loop)
    dim3 g4(HIDDEN / 256, BATCH);
    size_t shmem = (size_t)(TSTEPS * TSTEPS + TSTEPS * VSTRIDE) * sizeof(float);
    value_scan<<<g4, 256, shmem, stream>>>(ws_i, ws_C, out_vals, ws_vtr);

    // 5) mem = ETA * Vtr^T * Ktr (fp32 WMMA) -> 75 MB output at streaming rate
    dim3 g5(HIDDEN / 128, HIDDEN / 16, BATCH);
    mem_wmma<<<g5, 256, 0, stream>>>(ws_vtr, ws_ktr, out_mem);
}